// Decoder_22402549416573
// MI455X (gfx1250) — compile-verified
//
#include <hip/hip_runtime.h>
#include <hip/hip_bf16.h>

#define N_NODES 20000
#define N_EDGES 320000

typedef __attribute__((ext_vector_type(2))) float v2f;
typedef __attribute__((ext_vector_type(8))) float v8f;

// ---------------------------------------------------------------- utilities
__global__ void zero_i32(int* __restrict__ p, int n) {
  int i = blockIdx.x * blockDim.x + threadIdx.x;
  if (i < n) p[i] = 0;
}

__global__ void deg_count(const int* __restrict__ dst, int* __restrict__ deg, int E) {
  int i = blockIdx.x * blockDim.x + threadIdx.x;
  if (i < E) atomicAdd(&deg[dst[i]], 1);
}

// single-block exclusive scan (chunks of 1024) writing rowptr and cursor;
// deg may alias cursor (read-before-write per index within a chunk).
__global__ void scan_rowptr(const int* deg, int* rowptr, int* cursor, int n) {
  __shared__ int buf[1024];
  __shared__ int carry;
  int t = threadIdx.x;
  if (t == 0) carry = 0;
  __syncthreads();
  for (int base = 0; base < n; base += 1024) {
    int i = base + t;
    int v = (i < n) ? deg[i] : 0;
    buf[t] = v;
    __syncthreads();
    for (int off = 1; off < 1024; off <<= 1) {
      int add = (t >= off) ? buf[t - off] : 0;
      __syncthreads();
      buf[t] += add;
      __syncthreads();
    }
    int excl = carry + buf[t] - v;
    if (i < n) { rowptr[i] = excl; cursor[i] = excl; }
    __syncthreads();
    if (t == 0) carry += buf[1023];
    __syncthreads();
  }
  if (t == 0) rowptr[n] = carry;
}

__global__ void csr_fill(const int* __restrict__ src, const int* __restrict__ dst,
                         int* __restrict__ cursor, int* __restrict__ csr_src, int E) {
  int e = blockIdx.x * blockDim.x + threadIdx.x;
  if (e < E) {
    int pos = atomicAdd(&cursor[dst[e]], 1);
    csr_src[pos] = src[e];
  }
}

// ------------------------------------------------------------ WMMA f32 GEMM
// C[M x NCOLS] = A[M x K] * B[K x NCOLS], row-major, NCOLS compile-time.
// One wave owns MT consecutive 16-row M tiles for one 16-col N tile; each
// strided B pair feeds MT V_WMMA_F32_16X16X4_F32 ops. Software-pipelined:
// operands for k-step k+4 load while k-step k's WMMAs execute. All A tiles
// are indexed off one base pointer with compile-time strides (i*16*K floats)
// so loads stay in the GLOBAL address space with immediate offsets (no flat).
// EXEC is all-ones (tile counts divide exactly) as WMMA requires.
template <int K, int MT, int NCOLS>
__global__ __launch_bounds__(256) void wmma_gemm_f32(
    const float* __restrict__ A, const float* __restrict__ B,
    float* __restrict__ C, int M) {
  int wave = (blockIdx.x * blockDim.x + threadIdx.x) >> 5;
  int lane = threadIdx.x & 31;
  constexpr int ntiles = NCOLS >> 4;
  int tm0 = (wave / ntiles) * MT;
  int tn = wave % ntiles;
  if (tm0 * 16 >= M) return;  // wave-uniform
  int nm = lane & 15;         // M index for A, N index for B/C
  int half = lane >> 4;       // selects K pair (A/B) and M half (C)

  const float* ab = A + (size_t)(tm0 * 16 + nm) * K + half * 2;
  const float* bp = B + (size_t)(half * 2) * NCOLS + tn * 16 + nm;

  v8f acc[MT];
#pragma unroll
  for (int i = 0; i < MT; ++i) acc[i] = (v8f){};

  // prologue: k = 0 operands
  v2f b0, a0[MT];
  b0.x = bp[0];
  b0.y = bp[NCOLS];
#pragma unroll
  for (int i = 0; i < MT; ++i) {
    a0[i].x = ab[(size_t)i * 16 * K];
    a0[i].y = ab[(size_t)i * 16 * K + 1];
  }

#pragma unroll 4
  for (int k = 0; k < K; k += 4) {
    v2f b1, a1[MT];
    if (k + 4 < K) {  // load next k-step's operands into second buffer
      size_t kk = (size_t)(k + 4);
      b1.x = bp[kk * NCOLS];
      b1.y = bp[kk * NCOLS + NCOLS];
#pragma unroll
      for (int i = 0; i < MT; ++i) {
        a1[i].x = ab[(size_t)i * 16 * K + kk];
        a1[i].y = ab[(size_t)i * 16 * K + kk + 1];
      }
    }
#pragma unroll
    for (int i = 0; i < MT; ++i)
      acc[i] = __builtin_amdgcn_wmma_f32_16x16x4_f32(
          false, a0[i], false, b0, (short)0, acc[i], false, false);
    if (k + 4 < K) {  // rotate (renamed away after unrolling)
      b0 = b1;
#pragma unroll
      for (int i = 0; i < MT; ++i) a0[i] = a1[i];
    }
  }
#pragma unroll
  for (int i = 0; i < MT; ++i) {
    float* Cp = C + (size_t)((tm0 + i) * 16) * NCOLS + tn * 16 + nm;
#pragma unroll
    for (int r = 0; r < 8; ++r)
      Cp[(size_t)(half * 8 + r) * NCOLS] = acc[i][r];
  }
}

// ------------------------------------------------- attention dot products
// One wave per (node,head) row: el = feat·attn_l, er = feat·attn_r.
__global__ void attn_dots(const float* __restrict__ feat,
                          const float* __restrict__ attn_l,
                          const float* __restrict__ attn_r,
                          float* __restrict__ el, float* __restrict__ er,
                          int nrows, int H, int D) {
  int wid = (blockIdx.x * blockDim.x + threadIdx.x) >> 5;
  int lane = threadIdx.x & 31;
  if (wid >= nrows) return;
  int h = wid % H;
  const float* f = feat + (size_t)wid * D;
  const float* al = attn_l + h * D;
  const float* ar = attn_r + h * D;
  float sl = 0.f, sr = 0.f;
  for (int d = lane; d < D; d += 32) {
    float v = f[d];
    sl += v * al[d];
    sr += v * ar[d];
  }
#pragma unroll
  for (int off = 16; off; off >>= 1) {
    sl += __shfl_xor(sl, off, 32);
    sr += __shfl_xor(sr, off, 32);
  }
  if (lane == 0) { el[wid] = sl; er[wid] = sr; }
}

// ------------------------------------- fused edge softmax + aggregation
// One block (256 threads) per destination node. H*D == 512 in both layers:
// thread t owns features f0 = t and f1 = t + 256.
// MODE 0: layer 1 epilogue  -> out[v*256+t] = mean over 2 heads (+bias)
// MODE 1: layer 2 epilogue  -> relu + sigmoid written to out / out2 (N x 512)
template <int H, int D, int MODE>
__global__ __launch_bounds__(256) void gat_agg(
    const int* __restrict__ rowptr, const int* __restrict__ csr_src,
    const float* __restrict__ el, const float* __restrict__ er,
    const float* __restrict__ feat, const float* __restrict__ bias,
    float* __restrict__ out, float* __restrict__ out2) {
  constexpr int HD = H * D;  // 512
  int v = blockIdx.x;
  int t = threadIdx.x;
  int start = rowptr[v], end = rowptr[v + 1];

  __shared__ float red[256];
  __shared__ float mx[H], denom[H];
  __shared__ float alph[64 * H];
  __shared__ int usrc[64];

  // per-head max and exp-sum (leaky_relu slope 0.2 before softmax)
  for (int h = 0; h < H; ++h) {
    float erv = er[v * H + h];
    float lmax = -3.402823466e38f;
    for (int j = start + t; j < end; j += 256) {
      int u = csr_src[j];
      float e = el[u * H + h] + erv;
      e = (e > 0.f) ? e : 0.2f * e;
      lmax = fmaxf(lmax, e);
    }
    red[t] = lmax;
    __syncthreads();
    for (int off = 128; off; off >>= 1) {
      if (t < off) red[t] = fmaxf(red[t], red[t + off]);
      __syncthreads();
    }
    if (t == 0) mx[h] = red[0];
    __syncthreads();
    float m = mx[h];
    float ls = 0.f;
    for (int j = start + t; j < end; j += 256) {
      int u = csr_src[j];
      float e = el[u * H + h] + erv;
      e = (e > 0.f) ? e : 0.2f * e;
      ls += __expf(e - m);
    }
    red[t] = ls;
    __syncthreads();
    for (int off = 128; off; off >>= 1) {
      if (t < off) red[t] += red[t + off];
      __syncthreads();
    }
    if (t == 0) denom[h] = red[0];
    __syncthreads();
  }

  // accumulate alpha-weighted source features, 64 edges per LDS chunk;
  // prefetch next chunk's feature rows (global_prefetch_b8) behind the FMAs.
  float acc0 = 0.f, acc1 = 0.f;
  for (int base = start; base < end; base += 64) {
    int cnt = min(64, end - base);
    __syncthreads();
    if (t < cnt) usrc[t] = csr_src[base + t];
    if (t < H * 64) {
      int hh = t >> 6;
      int jj = t & 63;
      if (jj < cnt) {
        int u = csr_src[base + jj];
        float e = el[u * H + hh] + er[v * H + hh];
        e = (e > 0.f) ? e : 0.2f * e;
        alph[t] = __expf(e - mx[hh]) / denom[hh];
      }
    }
    __syncthreads();

    int nbase = base + 64;
    if (nbase < end) {
      int rem = end - nbase;
      int pj = t >> 2;  // 4 threads cover one upcoming row (4 x 128 floats)
      if (pj < rem && pj < 64) {
        int u = csr_src[nbase + pj];
        __builtin_prefetch(feat + (size_t)u * HD + (t & 3) * 128, 0, 3);
      }
    }

    constexpr int h1 = 256 / D;  // D=256 -> head 1 ; D=512 -> head 0
    for (int jj = 0; jj < cnt; ++jj) {
      const float* fu = feat + (size_t)usrc[jj] * HD;
      float a0 = alph[jj];
      float a1 = alph[h1 * 64 + jj];
      acc0 += a0 * fu[t];
      acc1 += a1 * fu[t + 256];
    }
  }

  if (MODE == 0) {
    // mean over 2 heads, + bias per (head, feature)
    out[(size_t)v * 256 + t] =
        0.5f * ((acc0 + bias[t]) + (acc1 + bias[256 + t]));
  } else {
    float r0 = fmaxf(acc0 + bias[t], 0.f);
    float r1 = fmaxf(acc1 + bias[t + 256], 0.f);
    size_t o = (size_t)v * 512 + t;
    out[o] = r0;
    out[o + 256] = r1;
    out2[o] = 1.f / (1.f + __expf(-r0));
    out2[o + 256] = 1.f / (1.f + __expf(-r1));
  }
}

// --------------------------------------------- domain-specific BatchNorm
__global__ void dom_count(const int* __restrict__ y, int* __restrict__ cnt, int N) {
  __shared__ int c[4];
  if (threadIdx.x < 4) c[threadIdx.x] = 0;
  __syncthreads();
  for (int i = blockIdx.x * blockDim.x + threadIdx.x; i < N;
       i += gridDim.x * blockDim.x)
    atomicAdd(&c[y[i]], 1);
  __syncthreads();
  if (threadIdx.x < 4) atomicAdd(&cnt[threadIdx.x], c[threadIdx.x]);
}

// one block per column c: deterministic tree-reduced per-domain sum / sumsq
__global__ void bn_stats(const float* __restrict__ x, const int* __restrict__ y,
                         float* __restrict__ sums, int N) {
  int c = blockIdx.x;
  int t = threadIdx.x;
  float s[4] = {0, 0, 0, 0}, q[4] = {0, 0, 0, 0};
  for (int r = t; r < N; r += 256) {
    float val = x[(size_t)r * 256 + c];
    int d = y[r];
    s[d] += val;
    q[d] += val * val;
  }
  __shared__ float red[256];
  for (int d = 0; d < 4; ++d) {
    red[t] = s[d];
    __syncthreads();
    for (int off = 128; off; off >>= 1) {
      if (t < off) red[t] += red[t + off];
      __syncthreads();
    }
    if (t == 0) sums[d * 256 + c] = red[0];
    __syncthreads();
    red[t] = q[d];
    __syncthreads();
    for (int off = 128; off; off >>= 1) {
      if (t < off) red[t] += red[t + off];
      __syncthreads();
    }
    if (t == 0) sums[1024 + d * 256 + c] = red[0];
    __syncthreads();
  }
}

__global__ void bn_apply(const float* __restrict__ x, const int* __restrict__ y,
                         const float* __restrict__ sums, const int* __restrict__ cnt,
                         const float* __restrict__ gamma, const float* __restrict__ beta,
                         float* __restrict__ out, int N) {
  int i = blockIdx.x * blockDim.x + threadIdx.x;
  if (i >= N * 256) return;
  int r = i >> 8;
  int c = i & 255;
  int d = y[r];
  float cf = (float)cnt[d];
  float den = fmaxf(cf, 1.f);
  float mu = sums[d * 256 + c] / den;
  float var = fmaxf(sums[1024 + d * 256 + c] / den - mu * mu, 0.f);
  float vx = x[i];
  float norm = gamma[d * 256 + c] * (vx - mu) * rsqrtf(var + 1e-5f) + beta[d * 256 + c];
  float val = (cf > 1.f) ? norm : vx;
  out[i] = (val > 0.f) ? val : 0.01f * val;  // LeakyReLU(0.01)
}

// ----------------------------------------------------------------- driver
extern "C" void kernel_launch(void* const* d_in, const int* in_sizes, int n_in,
                              void* d_out, int out_size, void* d_ws, size_t ws_size,
                              hipStream_t stream) {
  const int N = N_NODES, E = N_EDGES;
  const float* x       = (const float*)d_in[0];
  const float* W1      = (const float*)d_in[1];
  const float* attn_l1 = (const float*)d_in[2];
  const float* attn_r1 = (const float*)d_in[3];
  const float* bias1   = (const float*)d_in[4];
  const float* gamma   = (const float*)d_in[5];
  const float* beta    = (const float*)d_in[6];
  const float* W2      = (const float*)d_in[7];
  const float* attn_l2 = (const float*)d_in[8];
  const float* attn_r2 = (const float*)d_in[9];
  const float* bias2   = (const float*)d_in[10];
  const int*   y       = (const int*)d_in[11];
  const int*   esrc    = (const int*)d_in[12];
  const int*   edst    = (const int*)d_in[13];
  float* out = (float*)d_out;

  char* w = (char*)d_ws;
  auto carve = [&](size_t bytes) -> void* {
    void* p = (void*)w;
    w += (bytes + 255) & ~(size_t)255;
    return p;
  };
  int*   rowptr  = (int*)carve((size_t)(N + 1) * 4);
  int*   cursor  = (int*)carve((size_t)N * 4);          // also degree temp
  int*   csr_src = (int*)carve((size_t)E * 4);
  float* el1     = (float*)carve((size_t)N * 2 * 4);
  float* er1     = (float*)carve((size_t)N * 2 * 4);
  float* el2     = (float*)carve((size_t)N * 4);
  float* er2     = (float*)carve((size_t)N * 4);
  float* feat    = (float*)carve((size_t)N * 512 * 4);  // reused by both GEMMs
  float* o1      = (float*)carve((size_t)N * 256 * 4);
  float* h       = (float*)carve((size_t)N * 256 * 4);
  float* bnsums  = (float*)carve(2048 * 4);
  int*   dcnt    = (int*)carve(4 * 4);

  // --- CSR by destination ---
  zero_i32<<<(N + 255) / 256, 256, 0, stream>>>(cursor, N);
  zero_i32<<<1, 256, 0, stream>>>(dcnt, 4);
  deg_count<<<(E + 255) / 256, 256, 0, stream>>>(edst, cursor, E);
  scan_rowptr<<<1, 1024, 0, stream>>>(cursor, rowptr, cursor, N);
  csr_fill<<<(E + 255) / 256, 256, 0, stream>>>(esrc, edst, cursor, csr_src, E);

  // 20000/16 = 1250 M tiles = 5 * 250 ; 512/16 = 32 N tiles
  const int gemm_blocks = ((N / 16 / 5) * (512 / 16)) / 8;  // 8 waves / block

  // --- layer 1: feat1 = x @ W1 (20000x32 * 32x512), WMMA f32 ---
  wmma_gemm_f32<32, 5, 512><<<gemm_blocks, 256, 0, stream>>>(x, W1, feat, N);
  attn_dots<<<(N * 2 * 32 + 255) / 256, 256, 0, stream>>>(feat, attn_l1, attn_r1,
                                                          el1, er1, N * 2, 2, 256);
  gat_agg<2, 256, 0><<<N, 256, 0, stream>>>(rowptr, csr_src, el1, er1, feat,
                                            bias1, o1, o1);

  // --- domain-specific BN + LeakyReLU(0.01) ---
  dom_count<<<(N + 255) / 256, 256, 0, stream>>>(y, dcnt, N);
  bn_stats<<<256, 256, 0, stream>>>(o1, y, bnsums, N);
  bn_apply<<<(N * 256 + 255) / 256, 256, 0, stream>>>(o1, y, bnsums, dcnt,
                                                      gamma, beta, h, N);

  // --- layer 2: feat2 = h @ W2 (20000x256 * 256x512), WMMA f32 ---
  wmma_gemm_f32<256, 5, 512><<<gemm_blocks, 256, 0, stream>>>(h, W2, feat, N);
  attn_dots<<<(N * 32 + 255) / 256, 256, 0, stream>>>(feat, attn_l2, attn_r2,
                                                      el2, er2, N, 1, 512);
  gat_agg<1, 512, 1><<<N, 256, 0, stream>>>(rowptr, csr_src, el2, er2, feat,
                                            bias2, out, out + (size_t)N * 512);
}